// HeteroGAT_23441931501776
// MI455X (gfx1250) — compile-verified
//
#include <hip/hip_runtime.h>
#include <math.h>

// ---------------------------------------------------------------------------
// HeteroGAT for MI455X (gfx1250, wave32).
// Dense node projections via V_WMMA_F32_16X16X4_F32 (templated K, hoisted B
// fragments, 4 row-blocks per wave); edge softmax via 3-pass atomic segment
// reduction over L2-resident tables.
// ---------------------------------------------------------------------------

typedef float v2f __attribute__((ext_vector_type(2)));
typedef float v8f __attribute__((ext_vector_type(8)));

#define HF 32           // hidden width (fixed by model)
#define NEG_SLOPE 0.2f

__device__ __forceinline__ float leaky(float x) {
  return x > 0.f ? x : NEG_SLOPE * x;
}
// Monotone float<->uint encoding so atomicMax(u32) implements float max.
__device__ __forceinline__ unsigned enc_f(float f) {
  unsigned b = __float_as_uint(f);
  return (b & 0x80000000u) ? ~b : (b | 0x80000000u);
}
__device__ __forceinline__ float dec_f(unsigned u) {
  unsigned b = (u & 0x80000000u) ? (u ^ 0x80000000u) : ~u;
  return __uint_as_float(b);
}

// u[d] = W[d,32] @ avec[32]   (tiny)
__global__ void k_compute_u(const float* __restrict__ W, const float* __restrict__ avec,
                            float* __restrict__ u, int d) {
  int i = blockIdx.x * blockDim.x + threadIdx.x;
  if (i < d) {
    float s = 0.f;
    for (int h = 0; h < HF; ++h) s += W[i * HF + h] * avec[h];
    u[i] = s;
  }
}

// a[i] = dot(x[i,:d], u)
__global__ void k_node_alpha(const float* __restrict__ x, const float* __restrict__ u,
                             float* __restrict__ a, int n, int d) {
  int i = blockIdx.x * blockDim.x + threadIdx.x;
  if (i >= n) return;
  const float* xr = x + (long long)i * d;
  float s = 0.f;
  for (int k = 0; k < d; ++k) s += xr[k] * u[k];
  a[i] = s;
}

__global__ void k_init_mz(unsigned* __restrict__ m, float* __restrict__ z, int n) {
  int i = blockIdx.x * blockDim.x + threadIdx.x;
  if (i < n) { m[i] = 0x007FFFFFu; /* enc(-inf) */ z[i] = 0.f; }
}

// out[i*32+h] = b1[h] (+ b2[h])
__global__ void k_init_out(float* __restrict__ out, const float* __restrict__ b1,
                           const float* __restrict__ b2, int n) {
  long long t = (long long)blockIdx.x * blockDim.x + threadIdx.x;
  if (t >= (long long)n * HF) return;
  int h = (int)(t & (HF - 1));
  float v = b1[h];
  if (b2) v += b2[h];
  out[t] = v;
}

__global__ void k_relu(float* __restrict__ x, long long n) {
  long long t = (long long)blockIdx.x * blockDim.x + threadIdx.x;
  if (t < n) x[t] = fmaxf(x[t], 0.f);
}

// OUT[N,32] = X[N,D] @ W[D,32] via V_WMMA_F32_16X16X4_F32.
// Requires N % 16 == 0 (all launches satisfy this: 400000, 100000).
// Each wave handles BLKS consecutive 16-row blocks, reusing the per-lane
// W fragments (hoisted into VGPRs) across blocks and both 16-col N-tiles.
// A 16x4 f32 layout: lane L holds row L%16, Ks {2*(L/16), 2*(L/16)+1}.
// C/D 16x16 f32:     lane L, vgpr r -> [row r + 8*(L/16)][col L%16].
template <int D>
__global__ void k_gemm_wmma(const float* __restrict__ X, const float* __restrict__ W,
                            float* __restrict__ OUT, int N) {
  constexpr int BLKS = 4;                 // 16-row blocks per wave
  constexpr int KS   = D / 4;             // WMMA K-steps
  int wave = blockIdx.x * (blockDim.x >> 5) + (threadIdx.x >> 5);
  int lane = threadIdx.x & 31;
  int half = lane >> 4;                   // 0 or 1 (K-half / row-half selector)
  int l16  = lane & 15;

  // Hoist B fragments: bf[t][ks] covers N-tile t, K-step ks, this lane's K-half.
  v2f bf[2][KS];
#pragma unroll
  for (int ks = 0; ks < KS; ++ks) {
    int kk = ks * 4 + half * 2;
    bf[0][ks].x = W[(kk + 0) * HF + l16];
    bf[0][ks].y = W[(kk + 1) * HF + l16];
    bf[1][ks].x = W[(kk + 0) * HF + 16 + l16];
    bf[1][ks].y = W[(kk + 1) * HF + 16 + l16];
  }

  int blk0 = wave * BLKS;
#pragma unroll 1
  for (int b = 0; b < BLKS; ++b) {
    int row0 = (blk0 + b) * 16;
    if (row0 >= N) break;                 // wave-uniform: EXEC stays full for WMMA
    const float* xrow = X + (long long)(row0 + l16) * D + half * 2;
    v8f acc0 = {0.f, 0.f, 0.f, 0.f, 0.f, 0.f, 0.f, 0.f};
    v8f acc1 = acc0;
#pragma unroll
    for (int ks = 0; ks < KS; ++ks) {
      v2f a = *(const v2f*)(xrow + ks * 4);   // 8B-aligned: row*D and ks*4 even
      acc0 = __builtin_amdgcn_wmma_f32_16x16x4_f32(false, a, false, bf[0][ks],
                                                   (short)0, acc0, false, false);
      acc1 = __builtin_amdgcn_wmma_f32_16x16x4_f32(false, a, false, bf[1][ks],
                                                   (short)0, acc1, false, false);
    }
    float* orow = OUT + (long long)(row0 + half * 8) * HF;
#pragma unroll
    for (int r = 0; r < 8; ++r) {
      orow[r * HF + l16]      = acc0[r];
      orow[r * HF + 16 + l16] = acc1[r];
    }
  }
}

// pass 1: per-dst max of edge scores
__global__ void k_edge_max(const int* __restrict__ src, const int* __restrict__ dst,
                           const float* __restrict__ a_s, const float* __restrict__ a_d,
                           unsigned* __restrict__ m, int E) {
  int e = blockIdx.x * blockDim.x + threadIdx.x;
  if (e >= E) return;
  int dd = dst[e];
  float ev = leaky(a_s[src[e]] + a_d[dd]);
  atomicMax(&m[dd], enc_f(ev));
}

// pass 2: per-dst sum of exp(e - m)
__global__ void k_edge_sum(const int* __restrict__ src, const int* __restrict__ dst,
                           const float* __restrict__ a_s, const float* __restrict__ a_d,
                           const unsigned* __restrict__ m, float* __restrict__ z, int E) {
  int e = blockIdx.x * blockDim.x + threadIdx.x;
  if (e >= E) return;
  int dd = dst[e];
  float ev = leaky(a_s[src[e]] + a_d[dd]);
  atomicAdd(&z[dd], __expf(ev - dec_f(m[dd])));
}

// pass 3: out[dst] += attn * hs[src]; one edge per wave, one feature per lane
// (coalesced 128B hs gather + coalesced atomic burst).
__global__ void k_edge_msg(const int* __restrict__ src, const int* __restrict__ dst,
                           const float* __restrict__ a_s, const float* __restrict__ a_d,
                           const unsigned* __restrict__ m, const float* __restrict__ z,
                           const float* __restrict__ hs, float* __restrict__ out, int E) {
  long long t = (long long)blockIdx.x * blockDim.x + threadIdx.x;
  long long e = t >> 5;
  int f = (int)(t & 31);
  if (e >= E) return;
  int s  = src[e];
  int dd = dst[e];
  float ev   = leaky(a_s[s] + a_d[dd]);
  float attn = __expf(ev - dec_f(m[dd])) / (z[dd] + 1e-16f);
  atomicAdd(&out[(long long)dd * HF + f], hs[(long long)s * HF + f] * attn);
}

// final two linear layers: out = (relu(x)@W0 + b0)@W1 + b1
__global__ void k_head(const float* __restrict__ x, const float* __restrict__ W0,
                       const float* __restrict__ b0, const float* __restrict__ W1,
                       const float* __restrict__ b1, float* __restrict__ out,
                       int n, int dout) {
  int i = blockIdx.x * blockDim.x + threadIdx.x;
  if (i >= n) return;
  float xi[HF], h[HF];
#pragma unroll
  for (int k = 0; k < HF; ++k) xi[k] = fmaxf(x[(long long)i * HF + k], 0.f);
  for (int j = 0; j < HF; ++j) {
    float s = b0[j];
#pragma unroll
    for (int k = 0; k < HF; ++k) s += xi[k] * W0[k * HF + j];
    h[j] = s;
  }
  for (int j = 0; j < dout; ++j) {
    float s = b1[j];
#pragma unroll
    for (int k = 0; k < HF; ++k) s += h[k] * W1[k * dout + j];
    out[(long long)i * dout + j] = s;
  }
}

// ---------------------------------------------------------------------------

extern "C" void kernel_launch(void* const* d_in, const int* in_sizes, int n_in,
                              void* d_out, int out_size, void* d_ws, size_t ws_size,
                              hipStream_t stream) {
  (void)in_sizes; (void)n_in; (void)out_size; (void)ws_size;
  const int Nb = 400000, Ng = 100000, DIN = 16;
  const int E_BB = 4000000, E_GB = 1000000, E_BG = 1000000;

  const float* x_bus = (const float*)d_in[0];
  const float* x_gen = (const float*)d_in[1];
  const int* ei_bb = (const int*)d_in[2];
  const int* ei_gb = (const int*)d_in[3];
  const int* ei_bg = (const int*)d_in[4];

  // params pytree leaves (sorted keys): per GAT: Wdst, Wsrc, adst, asrc, b
  struct GatP { const float *Wdst, *Wsrc, *adst, *asrc, *b; };
  auto gat = [&](int base) {
    return GatP{ (const float*)d_in[base], (const float*)d_in[base + 1],
                 (const float*)d_in[base + 2], (const float*)d_in[base + 3],
                 (const float*)d_in[base + 4] };
  };
  GatP L0bb = gat(5),  L0bg = gat(10), L0gb = gat(15);
  GatP L1bb = gat(20), L1bg = gat(25), L1gb = gat(30);
  const float* lin0_bus_W = (const float*)d_in[35];
  const float* lin0_bus_b = (const float*)d_in[36];
  const float* lin0_gen_W = (const float*)d_in[37];
  const float* lin0_gen_b = (const float*)d_in[38];
  const float* lin1_bus_W = (const float*)d_in[39];
  const float* lin1_bus_b = (const float*)d_in[40];
  const float* lin1_gen_W = (const float*)d_in[41];
  const float* lin1_gen_b = (const float*)d_in[42];

  // ---- workspace carve-out (~186 MB) ----
  float* ws = (float*)d_ws;
  size_t off = 0;
  auto carve = [&](size_t nfl) { float* p = ws + off; off += nfl; return p; };
  float*    h_src = carve((size_t)Nb * HF);   // reused per edge type (src projection)
  float*    xb1   = carve((size_t)Nb * HF);   // layer-1 bus output
  float*    xg1   = carve((size_t)Ng * HF);   // layer-1 gen output
  float*    xb2   = carve((size_t)Nb * HF);   // layer-2 bus output
  float*    xg2   = carve((size_t)Ng * HF);   // layer-2 gen output
  float*    a_s   = carve(Nb);
  float*    a_d   = carve(Nb);
  unsigned* m_enc = (unsigned*)carve(Nb);
  float*    zacc  = carve(Nb);
  float*    u_s   = carve(HF);
  float*    u_d   = carve(HF);

  auto cdiv = [](long long a, long long b) { return (unsigned)((a + b - 1) / b); };

  auto run_gat = [&](const float* xs, int ns, const float* xd, int nd, int d,
                     const int* ei, int E, const GatP& p, float* out_acc) {
    k_compute_u<<<1, 32, 0, stream>>>(p.Wsrc, p.asrc, u_s, d);
    k_compute_u<<<1, 32, 0, stream>>>(p.Wdst, p.adst, u_d, d);
    // waves needed: each wave covers 64 rows; 8 waves per 256-thread block
    unsigned gemm_blocks = cdiv(cdiv(ns, 64), 8);
    if (d == 16) {
      k_gemm_wmma<16><<<gemm_blocks, 256, 0, stream>>>(xs, p.Wsrc, h_src, ns);
    } else {
      k_gemm_wmma<32><<<gemm_blocks, 256, 0, stream>>>(xs, p.Wsrc, h_src, ns);
    }
    k_node_alpha<<<cdiv(ns, 256), 256, 0, stream>>>(xs, u_s, a_s, ns, d);
    k_node_alpha<<<cdiv(nd, 256), 256, 0, stream>>>(xd, u_d, a_d, nd, d);
    k_init_mz<<<cdiv(nd, 256), 256, 0, stream>>>(m_enc, zacc, nd);
    const int* src = ei;
    const int* dst = ei + E;
    k_edge_max<<<cdiv(E, 256), 256, 0, stream>>>(src, dst, a_s, a_d, m_enc, E);
    k_edge_sum<<<cdiv(E, 256), 256, 0, stream>>>(src, dst, a_s, a_d, m_enc, zacc, E);
    k_edge_msg<<<cdiv((long long)E * 32, 256), 256, 0, stream>>>(
        src, dst, a_s, a_d, m_enc, zacc, h_src, out_acc, E);
  };

  // ---- layer 0 (d = 16) ----
  k_init_out<<<cdiv((long long)Nb * HF, 256), 256, 0, stream>>>(xb1, L0bb.b, L0gb.b, Nb);
  run_gat(x_bus, Nb, x_bus, Nb, DIN, ei_bb, E_BB, L0bb, xb1);
  run_gat(x_gen, Ng, x_bus, Nb, DIN, ei_gb, E_GB, L0gb, xb1);
  k_init_out<<<cdiv((long long)Ng * HF, 256), 256, 0, stream>>>(xg1, L0bg.b, nullptr, Ng);
  run_gat(x_bus, Nb, x_gen, Ng, DIN, ei_bg, E_BG, L0bg, xg1);
  k_relu<<<cdiv((long long)Nb * HF, 256), 256, 0, stream>>>(xb1, (long long)Nb * HF);
  k_relu<<<cdiv((long long)Ng * HF, 256), 256, 0, stream>>>(xg1, (long long)Ng * HF);

  // ---- layer 1 (d = 32) ----
  k_init_out<<<cdiv((long long)Nb * HF, 256), 256, 0, stream>>>(xb2, L1bb.b, L1gb.b, Nb);
  run_gat(xb1, Nb, xb1, Nb, HF, ei_bb, E_BB, L1bb, xb2);
  run_gat(xg1, Ng, xb1, Nb, HF, ei_gb, E_GB, L1gb, xb2);
  k_init_out<<<cdiv((long long)Ng * HF, 256), 256, 0, stream>>>(xg2, L1bg.b, nullptr, Ng);
  run_gat(xb1, Nb, xg1, Ng, HF, ei_bg, E_BG, L1bg, xg2);
  k_relu<<<cdiv((long long)Nb * HF, 256), 256, 0, stream>>>(xb2, (long long)Nb * HF);
  k_relu<<<cdiv((long long)Ng * HF, 256), 256, 0, stream>>>(xg2, (long long)Ng * HF);

  // ---- heads ----
  float* out_bus = (float*)d_out;
  float* out_gen = out_bus + (size_t)Nb * 4;
  k_head<<<cdiv(Nb, 128), 128, 0, stream>>>(xb2, lin0_bus_W, lin0_bus_b,
                                            lin1_bus_W, lin1_bus_b, out_bus, Nb, 4);
  k_head<<<cdiv(Ng, 128), 128, 0, stream>>>(xg2, lin0_gen_W, lin0_gen_b,
                                            lin1_gen_W, lin1_gen_b, out_gen, Ng, 2);
}